// TwoPhasePINNModel_41059887350379
// MI455X (gfx1250) — compile-verified
//
#include <hip/hip_runtime.h>
#include <math.h>

typedef __attribute__((ext_vector_type(2))) float v2f;
typedef __attribute__((ext_vector_type(4))) float v4f;
typedef __attribute__((ext_vector_type(8))) float v8f;

#define AS1 __attribute__((address_space(1)))
typedef AS1 const float* gcf;

#define NCH   7                    // jet channels: val, x, y, t, xx, yy, xy
#define FEAT  128
#define PAD   20                   // stride 20 words: conflict-free + 16B-aligned rows
#define HWORDS (NCH * FEAT * PAD)  // per-group activation-jet words (70 KB)
#define GROUPS_PER_WG 4            // 16-point groups per workgroup
#define PTS_PER_GROUP 16
#define NTW   4                    // N-tiles per wave (wave pair covers 8)

__device__ __forceinline__ float fast_tanh(float z) {
#if __has_builtin(__builtin_amdgcn_tanhf)
    return __builtin_amdgcn_tanhf(z);     // v_tanh_f32
#else
    return tanhf(z);
#endif
}

struct Ops { v2f B[NTW]; v2f A[NCH]; };

__device__ __forceinline__ void load_ops(Ops& o, gcf W, const float* H,
                                         int k, int half, int r, int ncol0) {
    gcf Wk = W + (k + 2 * half) * FEAT + ncol0 + r;
    #pragma unroll
    for (int nt = 0; nt < NTW; ++nt) {
        v2f bb;
        bb.x = Wk[nt * 16];
        bb.y = Wk[FEAT + nt * 16];
        o.B[nt] = bb;
    }
    const float* Hk = H + (k + 2 * half) * PAD + r;
    #pragma unroll
    for (int ch = 0; ch < NCH; ++ch) {
        v2f aa;
        aa.x = Hk[ch * FEAT * PAD];
        aa.y = Hk[ch * FEAT * PAD + PAD];
        o.A[ch] = aa;
    }
}

__device__ __forceinline__ void do_wmma(v8f (&acc)[NCH][NTW], const Ops& o) {
    #pragma unroll
    for (int ch = 0; ch < NCH; ++ch)
        #pragma unroll
        for (int nt = 0; nt < NTW; ++nt)
            acc[ch][nt] = __builtin_amdgcn_wmma_f32_16x16x4_f32(
                false, o.A[ch], false, o.B[nt],
                (short)0, acc[ch][nt], false, false);
}

__global__ __launch_bounds__(256, 1)
void pinn_twophase_kernel(const float* __restrict__ xin,
                          const float* __restrict__ yin,
                          const float* __restrict__ tin,
                          const float* __restrict__ W0p, const float* __restrict__ b0p,
                          const float* __restrict__ W1p, const float* __restrict__ b1p,
                          const float* __restrict__ W2p, const float* __restrict__ b2p,
                          const float* __restrict__ W3p, const float* __restrict__ b3p,
                          const float* __restrict__ W4p, const float* __restrict__ b4p,
                          float* __restrict__ out, int n)
{
    extern __shared__ float smem[];
    const int lane  = threadIdx.x & 31;
    const int wave  = threadIdx.x >> 5;
    const int group = wave >> 1;      // 16-point group (shares one H buffer)
    const int nh    = wave & 1;       // which half of the 8 N-tiles this wave owns
    const int r     = lane & 15;      // A-layout row (= point), B/D-layout column
    const int half  = lane >> 4;      // K sub-pair selector for A/B operands

    float* H = smem + group * HWORDS;     // H[ch][k][p] : ch*FEAT*PAD + k*PAD + p
    const int p0 = (blockIdx.x * GROUPS_PER_WG + group) * PTS_PER_GROUP;
    // block-uniform guard (barrier-safe); N is a multiple of 64 in this workload
    if ((blockIdx.x + 1) * GROUPS_PER_WG * PTS_PER_GROUP > n) return;

    gcf W0 = (gcf)W0p; gcf b0 = (gcf)b0p;
    gcf W4 = (gcf)W4p; gcf b4 = (gcf)b4p;
    gcf Ws[3] = {(gcf)W1p, (gcf)W2p, (gcf)W3p};
    gcf bs[3] = {(gcf)b1p, (gcf)b2p, (gcf)b3p};

    // ---------------- Layer 0: (x,y,t) -> 128, seed 2nd-order jet ----------------
    // the wave pair splits the group's fill
    for (int j = lane + 32 * nh; j < PTS_PER_GROUP * FEAT; j += 64) {
        int p  = j >> 7;
        int nf = j & 127;
        float xv = xin[p0 + p], yv = yin[p0 + p], tv = tin[p0 + p];
        float w0 = W0[nf], w1 = W0[FEAT + nf], w2 = W0[2 * FEAT + nf];
        float z  = fmaf(xv, w0, fmaf(yv, w1, fmaf(tv, w2, b0[nf])));
        float th = fast_tanh(z);
        float f1 = 1.0f - th * th;
        float f2 = -2.0f * th * f1;
        float* hp = H + nf * PAD + p;
        hp[0 * FEAT * PAD] = th;
        hp[1 * FEAT * PAD] = f1 * w0;
        hp[2 * FEAT * PAD] = f1 * w1;
        hp[3 * FEAT * PAD] = f1 * w2;
        hp[4 * FEAT * PAD] = f2 * w0 * w0;
        hp[5 * FEAT * PAD] = f2 * w1 * w1;
        hp[6 * FEAT * PAD] = f2 * w0 * w1;
    }
    __syncthreads();

    // ---------------- Hidden layers 1..3: 128 -> 128 with f32 WMMA ----------------
    for (int l = 0; l < 3; ++l) {
        gcf W = Ws[l];
        gcf b = bs[l];
        const int ncol0 = nh * (NTW * 16);     // this wave's output-column base

        v8f acc[NCH][NTW];
        #pragma unroll
        for (int nt = 0; nt < NTW; ++nt) {
            float bv = b[ncol0 + nt * 16 + r];  // bias only on value channel
            acc[0][nt] = (v8f){bv, bv, bv, bv, bv, bv, bv, bv};
            #pragma unroll
            for (int ch = 1; ch < NCH; ++ch)
                acc[ch][nt] = (v8f){0, 0, 0, 0, 0, 0, 0, 0};
        }

        // software-pipelined K loop: double-buffered A/B operands
        Ops op0, op1;
        load_ops(op0, W, H, 0, half, r, ncol0);
        #pragma unroll 1
        for (int k = 0; k < FEAT; k += 8) {
            load_ops(op1, W, H, k + 4, half, r, ncol0);
            do_wmma(acc, op0);
            if (k + 8 < FEAT)
                load_ops(op0, W, H, k + 8, half, r, ncol0);
            do_wmma(acc, op1);
        }
        __syncthreads();    // partner wave must finish reading H before overwrite

        // tanh jet (chain rule) in D-layout; vectorized in-place writeback
        #pragma unroll
        for (int nt = 0; nt < NTW; ++nt) {
            float hj[NCH][8];
            #pragma unroll
            for (int i = 0; i < 8; ++i) {
                float z   = acc[0][nt][i];
                float zx  = acc[1][nt][i], zy = acc[2][nt][i], zt = acc[3][nt][i];
                float zxx = acc[4][nt][i], zyy = acc[5][nt][i], zxy = acc[6][nt][i];
                float th = fast_tanh(z);
                float f1 = 1.0f - th * th;
                float f2 = -2.0f * th * f1;
                hj[0][i] = th;
                hj[1][i] = f1 * zx;
                hj[2][i] = f1 * zy;
                hj[3][i] = f1 * zt;
                hj[4][i] = fmaf(f2 * zx, zx, f1 * zxx);
                hj[5][i] = fmaf(f2 * zy, zy, f1 * zyy);
                hj[6][i] = fmaf(f2 * zx, zy, f1 * zxy);
            }
            // row (ncol0 + nt*16 + r) of next H, points 8*half..+7: 16B-aligned run
            int row = ncol0 + nt * 16 + r;
            #pragma unroll
            for (int ch = 0; ch < NCH; ++ch) {
                v4f* hp4 = (v4f*)(H + 4 * (ch * (FEAT * PAD / 4) +
                                           row * (PAD / 4) + 2 * half));
                hp4[0] = (v4f){hj[ch][0], hj[ch][1], hj[ch][2], hj[ch][3]};
                hp4[1] = (v4f){hj[ch][4], hj[ch][5], hj[ch][6], hj[ch][7]};
            }
        }
        __syncthreads();    // all columns written before next layer's reads
    }

    // ------- Layer 4 (128 -> 4, linear) + PDE: even wave of each pair only -------
    if (nh == 0) {
        v8f acc4[NCH];
        {
            float m  = (r < 4) ? 1.0f : 0.0f;
            float bv = b4[r & 3] * m;
            acc4[0] = (v8f){bv, bv, bv, bv, bv, bv, bv, bv};
            #pragma unroll
            for (int ch = 1; ch < NCH; ++ch)
                acc4[ch] = (v8f){0, 0, 0, 0, 0, 0, 0, 0};

            #pragma unroll 1
            for (int k = 0; k < FEAT; k += 4) {
                int k0 = k + 2 * half;
                v2f Bv;
                Bv.x = W4[k0 * 4 + (r & 3)] * m;
                Bv.y = W4[(k0 + 1) * 4 + (r & 3)] * m;
                v2f Av[NCH];
                const float* Hk = H + k0 * PAD + r;
                #pragma unroll
                for (int ch = 0; ch < NCH; ++ch) {
                    v2f aa;
                    aa.x = Hk[ch * FEAT * PAD];
                    aa.y = Hk[ch * FEAT * PAD + PAD];
                    Av[ch] = aa;
                }
                #pragma unroll
                for (int ch = 0; ch < NCH; ++ch)
                    acc4[ch] = __builtin_amdgcn_wmma_f32_16x16x4_f32(
                        false, Av[ch], false, Bv, (short)0, acc4[ch], false, false);
            }
        }

        // stage output jet to LDS: J[ch][o][p] (reuse H; this wave's reads are done)
        #pragma unroll
        for (int ch = 0; ch < NCH; ++ch)
            #pragma unroll
            for (int i = 0; i < 8; ++i) {
                int p = i + 8 * half;
                H[(ch * 16 + r) * PAD + p] = acc4[ch][i];
            }

        if (lane < 16) {
            const int p = lane;
            auto J = [&](int ch, int o) -> float { return H[(ch * 16 + o) * PAD + p]; };

            const float RHO1 = 1000.0f, RHO2 = 1.225f;
            const float MU1 = 0.001f, MU2 = 1.8e-5f;
            const float SIGMA = 0.0728f, GACC = 9.81f;

            float u = J(0, 0), v = J(0, 1), a = J(0, 3);
            float u_x = J(1, 0), u_y = J(2, 0), u_t = J(3, 0);
            float v_x = J(1, 1), v_y = J(2, 1), v_t = J(3, 1);
            float p_x = J(1, 2), p_y = J(2, 2);
            float a_x = J(1, 3), a_y = J(2, 3), a_t = J(3, 3);
            float u_xx = J(4, 0), u_yy = J(5, 0);
            float v_xx = J(4, 1), v_yy = J(5, 1);
            float a_xx = J(4, 3), a_yy = J(5, 3), a_xy = J(6, 3);

            float mu   = MU2 + (MU1 - MU2) * a;
            float mu_x = (MU1 - MU2) * a_x;
            float mu_y = (MU1 - MU2) * a_y;
            float rho  = RHO2 + (RHO1 - RHO2) * a;
            float g    = sqrtf(a_x * a_x + a_y * a_y + 1e-12f);
            float curv = -((a_xx + a_yy) / g
                         - (a_x * a_x * a_xx + a_y * a_y * a_yy + 2.0f * a_x * a_y * a_xy)
                           / (g * g * g));
            float one_Re   = mu / RHO2;
            float one_Re_x = mu_x / RHO2;
            float one_Re_y = mu_y / RHO2;
            float one_We   = SIGMA / RHO2;
            float one_Fr   = GACC;
            float rr = rho / RHO2;

            float PDE_m = u_x + v_y;
            float PDE_a = a_t + u * a_x + v * a_y;
            float PDE_u = (u_t + u * u_x + v * u_y) * rr + p_x - one_We * curv * a_x
                          - one_Re * (u_xx + u_yy) - 2.0f * one_Re_x * u_x
                          - one_Re_y * (u_y + v_x);
            float PDE_v = (v_t + u * v_x + v * v_y) * rr + p_y - one_We * curv * a_y
                          - rr * one_Fr
                          - one_Re * (v_xx + v_yy) - 2.0f * one_Re_y * v_y
                          - one_Re_x * (u_y + v_x);

            float* op = out + (size_t)(p0 + p) * 4;
            op[0] = PDE_m;
            op[1] = PDE_u;
            op[2] = PDE_v;
            op[3] = PDE_a;
        }
    }
}

extern "C" void kernel_launch(void* const* d_in, const int* in_sizes, int n_in,
                              void* d_out, int out_size, void* d_ws, size_t ws_size,
                              hipStream_t stream) {
    (void)n_in; (void)d_ws; (void)ws_size; (void)out_size;
    const float* x  = (const float*)d_in[0];
    const float* y  = (const float*)d_in[1];
    const float* t  = (const float*)d_in[2];
    const float* W0 = (const float*)d_in[3];
    const float* b0 = (const float*)d_in[4];
    const float* W1 = (const float*)d_in[5];
    const float* b1 = (const float*)d_in[6];
    const float* W2 = (const float*)d_in[7];
    const float* b2 = (const float*)d_in[8];
    const float* W3 = (const float*)d_in[9];
    const float* b3 = (const float*)d_in[10];
    const float* W4 = (const float*)d_in[11];
    const float* b4 = (const float*)d_in[12];
    float* out = (float*)d_out;

    const int n   = in_sizes[0];
    const int ppw = GROUPS_PER_WG * PTS_PER_GROUP;        // 64 points per workgroup
    const int wgs = (n + ppw - 1) / ppw;
    const size_t shmem = (size_t)GROUPS_PER_WG * HWORDS * sizeof(float); // ~287 KB / WGP

    hipFuncSetAttribute((const void*)pinn_twophase_kernel,
                        hipFuncAttributeMaxDynamicSharedMemorySize, (int)shmem);

    pinn_twophase_kernel<<<wgs, 256, shmem, stream>>>(
        x, y, t, W0, b0, W1, b1, W2, b2, W3, b3, W4, b4, out, n);
}